// QuantumFeatureExtractor_83863531421742
// MI455X (gfx1250) — compile-verified
//
#include <hip/hip_runtime.h>
#include <math.h>

// CDNA5 / gfx1250 quantum feature extractor (round 2).
// One workgroup per batch element; 2^14 complex64 state resident in LDS
// (128 KB of 320 KB WGP pool; 2 WGs co-resident). Gates via
// V_WMMA_F32_16X16X4_F32 with the realified 4x4 gate in A rows 0..3 and 16
// amplitude pairs per WMMA in B columns. RX*RY*RZ fused per qubit (U=RZ*RY*RX),
// chunk loop unrolled x2 for two ds_load_b64 in flight per wave.

typedef __attribute__((ext_vector_type(2))) float v2f;
typedef __attribute__((ext_vector_type(8))) float v8f;

#define NQ       14
#define DIM      16384        // 2^14
#define NPARAMS  140
#define BATCH    64
#define NTHREADS 512
#define NWAVES   (NTHREADS / 32)
#define TWO_PI_F 6.28318530717958647692f

// shared layout (floats): psi[2*DIM] | ang[144] | acc[4]
#define SMEM_FLOATS (2 * DIM + 144 + 4)
#define SMEM_BYTES  (SMEM_FLOATS * sizeof(float))

__device__ __forceinline__ int insert_zero(int v, int b) {
    return ((v >> b) << (b + 1)) | (v & ((1 << b) - 1));
}

__global__ __launch_bounds__(NTHREADS)
void qfe_simulate_kernel(const float* __restrict__ params,
                         const float* __restrict__ base_params,
                         float* __restrict__ out) {
    const int b   = blockIdx.x;
    const int tid = threadIdx.x;

    extern __shared__ float smem[];
    float2* psi = (float2*)smem;             // 16384 complex amplitudes
    float*  ang = smem + 2 * DIM;            // 140 angles
    float*  acc = ang + 144;                 // 3 reduction accumulators

    // ---- init state |0...0> and angles = sigmoid(params + base) * 2pi ----
    for (int z = tid; z < DIM; z += NTHREADS) psi[z] = make_float2(0.f, 0.f);
    for (int k = tid; k < NPARAMS; k += NTHREADS) {
        float x = params[b * NPARAMS + k] + base_params[k];
        ang[k] = (1.0f / (1.0f + expf(-x))) * TWO_PI_F;
    }
    if (tid == 0) psi[0] = make_float2(1.f, 0.f);
    __syncthreads();

    const int lane = tid & 31;   // wave32
    const int wave = tid >> 5;
    const int m    = lane & 15;  // M index (A) / N index (B,D)
    const int hi   = lane >> 4;  // 0: K=0,1 half ; 1: K=2,3 half

    // ---- 84 gate passes (14 fused 1q + 28 CRX per layer, 2 layers) ----
    for (int gs = 0; gs < 84; ++gs) {
        const int layer = gs / 42;
        const int st    = gs - layer * 42;
        const int abase = layer * 70;

        float u00r, u00i, u01r, u01i, u10r, u10i, u11r, u11i;
        int bt, bc;
        if (st < 14) {
            // fused U = RZ(a3) * RY(a2) * RX(a1) on qubit q
            const int q = st;
            float a1 = 0.5f * ang[abase + 3 * q];
            float a2 = 0.5f * ang[abase + 3 * q + 1];
            float a3 = 0.5f * ang[abase + 3 * q + 2];
            float c1 = cosf(a1), s1 = sinf(a1);
            float c2 = cosf(a2), s2 = sinf(a2);
            float c3 = cosf(a3), s3 = sinf(a3);
            u00r =  c3 * c2 * c1 + s3 * s2 * s1;
            u00i =  c3 * s2 * s1 - s3 * c2 * c1;
            u01r = -c3 * s2 * c1 - s3 * c2 * s1;
            u01i = -c3 * c2 * s1 + s3 * s2 * c1;
            u10r =  c3 * s2 * c1 + s3 * c2 * s1;
            u10i =  s3 * s2 * c1 - c3 * c2 * s1;
            u11r =  c3 * c2 * c1 + s3 * s2 * s1;
            u11i =  s3 * c2 * c1 - c3 * s2 * s1;
            bt = 13 - q; bc = -1;
        } else {
            int ctrl, tgt, aidx;
            if (st < 28) { int i = st - 14;           ctrl = i; tgt = (i + 1)  % NQ; aidx = abase + 42 + i; }
            else         { int k = st - 28; int i = 13 - k; ctrl = i; tgt = (i + 13) % NQ; aidx = abase + 56 + k; }
            float th = 0.5f * ang[aidx];
            float c = cosf(th), s = sinf(th);
            u00r = c;   u00i = 0.f;
            u01r = 0.f; u01i = -s;
            u10r = 0.f; u10i = -s;
            u11r = c;   u11i = 0.f;
            bt = 13 - tgt; bc = 13 - ctrl;
        }

        // Per-lane A-matrix entries (16x4 f32 layout): lane holds
        // A[m][2*hi], A[m][2*hi+1]; realified rows: even m -> (uR, -uI),
        // odd m -> (uI, uR); rows 4..15 are zero.
        float uR = (m & 2) ? (hi ? u11r : u10r) : (hi ? u01r : u00r);
        float uI = (m & 2) ? (hi ? u11i : u10i) : (hi ? u01i : u00i);
        v2f a;
        a.x = (m & 1) ? uI : uR;
        a.y = (m & 1) ? uR : -uI;
        if (m >= 4) { a.x = 0.f; a.y = 0.f; }

        const int chunks = (bc >= 0) ? (4096 >> 4) : (8192 >> 4);
        int blo = 0, bhi2 = 0;
        if (bc >= 0) { blo = (bc < bt) ? bc : bt; bhi2 = (bc < bt) ? bt : bc; }

        // Unroll x2: two independent ds_load_b64 in flight per iteration.
        for (int ch = 2 * wave; ch < chunks; ch += 2 * NWAVES) {
            const int p0 = (ch << 4) | m;
            const int p1 = p0 + 16;
            int i0a, i0b;
            if (bc >= 0) {
                int t = insert_zero(insert_zero(p0, blo), bhi2);
                i0a = t | (1 << bc);
                t = insert_zero(insert_zero(p1, blo), bhi2);
                i0b = t | (1 << bc);
            } else {
                i0a = insert_zero(p0, bt);
                i0b = insert_zero(p1, bt);
            }
            const int i1a = i0a | (1 << bt);
            const int i1b = i0b | (1 << bt);

            // B (4x16 f32): lanes 0-15 carry (r0,i0)=K0,K1; lanes 16-31 (r1,i1)=K2,K3.
            float2 bva = psi[hi ? i1a : i0a];
            float2 bvb = psi[hi ? i1b : i0b];
            v2f ba;  ba.x  = bva.x; ba.y  = bva.y;
            v2f bb2; bb2.x = bvb.x; bb2.y = bvb.y;

            v8f cz = {};
            v8f da = __builtin_amdgcn_wmma_f32_16x16x4_f32(
                false, a, false, ba, (short)0, cz, false, false);
            v8f db = __builtin_amdgcn_wmma_f32_16x16x4_f32(
                false, a, false, bb2, (short)0, cz, false, false);

            if (lane < 16) {
                psi[i0a] = make_float2(da[0], da[1]);
                psi[i1a] = make_float2(da[2], da[3]);
                psi[i0b] = make_float2(db[0], db[1]);
                psi[i1b] = make_float2(db[2], db[3]);
            }
        }
        __syncthreads();
    }

    // ---- expectation values: per wire w, X/Y/Z from amplitude pairs ----
    for (int w = 0; w < NQ; ++w) {
        const int bw = 13 - w;
        if (tid == 0) { acc[0] = 0.f; acc[1] = 0.f; acc[2] = 0.f; }
        __syncthreads();

        float sx = 0.f, sy = 0.f, sz = 0.f;
        for (int p = tid; p < (DIM >> 1); p += NTHREADS) {
            const int z0 = insert_zero(p, bw);
            const int z1 = z0 | (1 << bw);
            float2 A2 = psi[z0];
            float2 B2 = psi[z1];
            sx += 2.0f * (A2.x * B2.x + A2.y * B2.y);
            sy += 2.0f * (A2.x * B2.y - A2.y * B2.x);
            sz += (A2.x * A2.x + A2.y * A2.y) - (B2.x * B2.x + B2.y * B2.y);
        }
        #pragma unroll
        for (int off = 16; off > 0; off >>= 1) {
            sx += __shfl_down(sx, off);
            sy += __shfl_down(sy, off);
            sz += __shfl_down(sz, off);
        }
        if (lane == 0) {
            atomicAdd(&acc[0], sx);   // ds_add_f32
            atomicAdd(&acc[1], sy);
            atomicAdd(&acc[2], sz);
        }
        __syncthreads();
        if (tid == 0) {
            out[b * 42 + w]      = acc[0];  // X
            out[b * 42 + 14 + w] = acc[1];  // Y
            out[b * 42 + 28 + w] = acc[2];  // Z
        }
    }
}

extern "C" void kernel_launch(void* const* d_in, const int* in_sizes, int n_in,
                              void* d_out, int out_size, void* d_ws, size_t ws_size,
                              hipStream_t stream) {
    (void)in_sizes; (void)n_in; (void)out_size; (void)d_ws; (void)ws_size;
    const float* params      = (const float*)d_in[0];  // [64,140] f32
    const float* base_params = (const float*)d_in[1];  // [140]   f32
    float* out = (float*)d_out;                        // [64,42] f32

    qfe_simulate_kernel<<<dim3(BATCH), dim3(NTHREADS), SMEM_BYTES, stream>>>(
        params, base_params, out);
}